// LinearExponentialDelayed_38491496907209
// MI455X (gfx1250) — compile-verified
//
#include <hip/hip_runtime.h>

// Problem constants from the reference
#define B_DIM   16
#define OUT_DIM 1024
#define IN_DIM  2048
#define NROWS   (B_DIM * OUT_DIM)   // 16384 (b,o) rows
#define NPART   16                  // partial sums per row (feeds one WMMA tile)

typedef __attribute__((ext_vector_type(2))) float v2f;
typedef __attribute__((ext_vector_type(4))) float v4f;
typedef __attribute__((ext_vector_type(8))) float v8f;
typedef __attribute__((ext_vector_type(4))) int   v4i;

// ---------------------------------------------------------------------------
// Kernel 1: fused streaming pass. One block (256 threads) per (b,o) row.
//   - trace_new = trace + (dt/tau)*(alpha*x - trace)   -> folded output
//   - spike     = (delay + delay_init*x == 1.0)
//   - partial pot reduction: acc += spike ? W[o,i] : 0, reduced to 16
//     partials per row (2 half-wave sums per each of the 8 waves).
//
// Cache policy (MI455X, 192MB L2): trace/delay/delay_init/folded are
// single-use streams -> nontemporal (TH=NT) so they don't evict the 16x-reused
// 8MB weight matrix; weight/x stay on the regular (RT) path.
// ---------------------------------------------------------------------------
__global__ __launch_bounds__(256) void led_stream_kernel(
    const int*   __restrict__ x,       // (B,1,IN) int32
    const float* __restrict__ W,       // (OUT,IN)
    const float* __restrict__ trace,   // (B,OUT,IN)
    const float* __restrict__ delay,   // (B,OUT,IN)
    const float* __restrict__ dinit,   // (B,OUT,IN)
    const float* __restrict__ dt_p,
    const float* __restrict__ tau_p,
    const float* __restrict__ alpha_p,
    float*       __restrict__ folded,  // (B,OUT,IN) flat (reshape is a view)
    float*       __restrict__ partials)// (NROWS, NPART) workspace
{
    const int row = blockIdx.x;            // row = b*OUT + o
    const int b   = row >> 10;             // OUT_DIM = 1024
    const int o   = row & (OUT_DIM - 1);

    const float k     = dt_p[0] / tau_p[0];
    const float alpha = alpha_p[0];

    const size_t base = (size_t)row * IN_DIM;
    const v4f* tr4 = (const v4f*)(trace + base);
    const v4f* dl4 = (const v4f*)(delay + base);
    const v4f* di4 = (const v4f*)(dinit + base);
    const v4f* w4  = (const v4f*)(W + (size_t)o * IN_DIM);
    const v4i* x4  = (const v4i*)(x + (size_t)b * IN_DIM);
    v4f*       out4 = (v4f*)(folded + base);

    float acc = 0.0f;

    // IN_DIM/4 = 512 float4 chunks over 256 threads -> 2 iterations
    #pragma unroll
    for (int it = 0; it < (IN_DIM / 4) / 256; ++it) {
        const int c = it * 256 + threadIdx.x;

        if (it == 0) {
            // gfx1250 global_prefetch_b8 for the second trip
            __builtin_prefetch((const void*)(tr4 + c + 256), 0, 3);
            __builtin_prefetch((const void*)(dl4 + c + 256), 0, 3);
            __builtin_prefetch((const void*)(di4 + c + 256), 0, 3);
        }

        // Single-use streams: nontemporal loads (TH=NT)
        v4f t  = __builtin_nontemporal_load(tr4 + c);
        v4f d  = __builtin_nontemporal_load(dl4 + c);
        v4f di = __builtin_nontemporal_load(di4 + c);
        // Reused data: regular loads (L2-resident)
        v4i xi = x4[c];
        v4f w  = w4[c];

        float xf;
        xf  = (float)xi.x;
        t.x = fmaf(k, fmaf(alpha, xf, -t.x), t.x);
        acc += ((d.x + di.x * xf) == 1.0f) ? w.x : 0.0f;

        xf  = (float)xi.y;
        t.y = fmaf(k, fmaf(alpha, xf, -t.y), t.y);
        acc += ((d.y + di.y * xf) == 1.0f) ? w.y : 0.0f;

        xf  = (float)xi.z;
        t.z = fmaf(k, fmaf(alpha, xf, -t.z), t.z);
        acc += ((d.z + di.z * xf) == 1.0f) ? w.z : 0.0f;

        xf  = (float)xi.w;
        t.w = fmaf(k, fmaf(alpha, xf, -t.w), t.w);
        acc += ((d.w + di.w * xf) == 1.0f) ? w.w : 0.0f;

        // Write-once output stream: nontemporal store
        __builtin_nontemporal_store(t, out4 + c);
    }

    // Half-wave (16-lane) butterfly reduction: lane 0 holds sum of lanes 0-15,
    // lane 16 holds sum of lanes 16-31. 8 waves * 2 halves = 16 partials/row.
    #pragma unroll
    for (int m = 1; m < 16; m <<= 1)
        acc += __shfl_xor(acc, m, 32);

    const int lane = threadIdx.x & 31;
    const int wv   = threadIdx.x >> 5;      // 0..7
    if ((lane & 15) == 0)
        partials[(size_t)row * NPART + (wv << 1) + (lane >> 4)] = acc;
}

// ---------------------------------------------------------------------------
// Kernel 2: WMMA row-sum reduction. One wave per 16 rows.
// A (16x4 f32) holds a K-slice of the 16x16 partials tile; B is all-ones, so
// D = A*B + C accumulates row sums over K. Four accumulating
// v_wmma_f32_16x16x4_f32 ops sum all 16 partials -> pot for 16 rows.
//
// A layout (32-bit A 16x4): lanes 0-15 hold {K=0,K=1}, lanes 16-31 hold
// {K=2,K=3}, row M = lane & 15 -> lane loads one aligned float2.
// D layout: VGPR r = row r (lanes 0-15) / row 8+r (lanes 16-31); every N
// column is identical (B==1), so lane 0 stores rows 0-7, lane 16 rows 8-15.
// ---------------------------------------------------------------------------
__global__ __launch_bounds__(128) void led_wmma_reduce_kernel(
    const float* __restrict__ partials,   // (NROWS, NPART)
    float*       __restrict__ pot)        // (NROWS) == (B,1,OUT) flat
{
    const int wave = blockIdx.x * (blockDim.x >> 5) + (threadIdx.x >> 5);
    const int lane = threadIdx.x & 31;
    const int row0 = wave * 16;           // grid sized exactly: always in range

    const int m  = row0 + (lane & 15);    // which row this lane feeds
    const int kh = (lane >> 4);           // 0: K-pair {0,1}, 1: K-pair {2,3}

    const v2f* p2 = (const v2f*)partials; // NPART/2 = 8 float2 per row

    v2f ones;
    ones.x = 1.0f;
    ones.y = 1.0f;

    v8f c = {};
    #pragma unroll
    for (int j = 0; j < 4; ++j) {
        v2f a = p2[(size_t)m * (NPART / 2) + j * 2 + kh];
        c = __builtin_amdgcn_wmma_f32_16x16x4_f32(
                /*neg_a=*/false, a, /*neg_b=*/false, ones,
                /*c_mod=*/(short)0, c, /*reuse_a=*/false, /*reuse_b=*/false);
    }

    if ((lane & 15) == 0) {
        float* dst = pot + row0 + (lane >> 4) * 8;   // rows 0-7 or 8-15
        #pragma unroll
        for (int r = 0; r < 8; ++r)
            dst[r] = c[r];
    }
}

// ---------------------------------------------------------------------------
// Launch: d_in order = {x, weight, trace, delay, delay_init, dt, tau_t, alpha_t}
// d_out = [pot (16384 f32) | folded (16*1024*2048 f32)]
// ---------------------------------------------------------------------------
extern "C" void kernel_launch(void* const* d_in, const int* in_sizes, int n_in,
                              void* d_out, int out_size, void* d_ws, size_t ws_size,
                              hipStream_t stream) {
    const int*   x     = (const int*)d_in[0];
    const float* W     = (const float*)d_in[1];
    const float* trace = (const float*)d_in[2];
    const float* delay = (const float*)d_in[3];
    const float* dinit = (const float*)d_in[4];
    const float* dt_p  = (const float*)d_in[5];
    const float* tau_p = (const float*)d_in[6];
    const float* al_p  = (const float*)d_in[7];

    float* pot      = (float*)d_out;                  // first B*OUT floats
    float* folded   = (float*)d_out + NROWS;          // remaining B*OUT*IN
    float* partials = (float*)d_ws;                   // NROWS*NPART floats (1 MB)

    led_stream_kernel<<<NROWS, 256, 0, stream>>>(
        x, W, trace, delay, dinit, dt_p, tau_p, al_p, folded, partials);

    // NROWS/16 = 1024 waves; 4 waves per 128-thread block -> 256 blocks (exact)
    led_wmma_reduce_kernel<<<(NROWS / 16) / 4, 128, 0, stream>>>(partials, pot);
}